// xModalKD_42975442764684
// MI455X (gfx1250) — compile-verified
//
#include <hip/hip_runtime.h>

#define NVOX 150000
#define NPTS 400000
#define MPTS 250000
#define HDIM 64
#define NCLS 20
#define NSC  4
#define NBINS 16384

typedef float v2f __attribute__((ext_vector_type(2)));
typedef float v8f __attribute__((ext_vector_type(8)));

// ---------------------------------------------------------------------------
// Tall-skinny GEMM: Out[r, ocol+c] = act( A'[r,:] @ W[:,c] + bias[c] )
//   A'[r,k] = A[g(r), k]  (k<64 / no concat) ; A2[g(r), k-64] (k>=64, CONCAT)
//   g(r) = gidx[r] if GATHER else r
//   ACT: 0=none, 1=relu, 2=fused gate: out = relu(gate_src[r,c]*sigmoid(acc))
// One wave computes a 16x32 tile (two 16x16 WMMA accumulators sharing the A
// fragment) via V_WMMA_F32_16X16X4_F32 (exact fp32 path).
// K/LDW/NOUT are compile-time: the K loop fully unrolls into straight-line
// global_load(+imm offset) / v_wmma code with zero per-step address math.
// Requires: Nrows % 16 == 0 and ceil(NOUT/16) even (NOUT=128/64/20 all OK).
// Out-of-range columns are index-clamped on the B load; their accumulator
// columns are garbage but never stored (WMMA columns are independent).
// ---------------------------------------------------------------------------
template<int K, int LDW, int NOUT, int ACT, bool CONCAT, bool GATHER>
__global__ __launch_bounds__(256) void gemm16_wmma(
    const float* __restrict__ A, int lda,
    const float* __restrict__ A2, int lda2,
    const int* __restrict__ gidx,
    const float* __restrict__ W,
    const float* __restrict__ bias,
    float* __restrict__ Out, int ldo, int ocol,
    const float* __restrict__ gate_src, int ldg,
    int Nrows)
{
  constexpr int COLPAIRS = ((NOUT + 15) / 16) / 2;
  int wave = (blockIdx.x * blockDim.x + threadIdx.x) >> 5;
  int lane = threadIdx.x & 31;
  int rowTiles = Nrows >> 4;
  if (wave >= rowTiles * COLPAIRS) return;    // uniform per-wave exit
  int rowTile = wave / COLPAIRS;
  int colPair = wave - rowTile * COLPAIRS;

  int rlane = lane & 15;
  int khalf = (lane >> 4) * 2;                // ISA 16x4 fp32 A/B K sub-index
  int arow  = rowTile * 16 + rlane;
  int col0  = colPair * 32 + rlane;           // always < NOUT for our shapes
  int col1  = col0 + 16;                      // may exceed NOUT (NOUT=20 case)
  int c1    = (col1 < NOUT) ? col1 : (NOUT - 1);  // clamp: garbage never stored

  int asrc = GATHER ? gidx[arow] : arow;
  const float* ap  = A + (long)asrc * lda + khalf;
  const float* wp0 = W + khalf * LDW + col0;
  const float* wp1 = W + khalf * LDW + c1;

  v8f acc0 = {}, acc1 = {};
  constexpr int K1 = CONCAT ? 64 : K;

#pragma unroll
  for (int k0 = 0; k0 < K1; k0 += 4) {
    v2f a = *(const v2f*)(ap + k0);
    v2f b0, b1;
    b0.x = wp0[k0 * LDW];       b0.y = wp0[k0 * LDW + LDW];
    b1.x = wp1[k0 * LDW];       b1.y = wp1[k0 * LDW + LDW];
    acc0 = __builtin_amdgcn_wmma_f32_16x16x4_f32(false, a, false, b0,
                                                 (short)0, acc0, false, false);
    acc1 = __builtin_amdgcn_wmma_f32_16x16x4_f32(false, a, false, b1,
                                                 (short)0, acc1, false, false);
  }
  if constexpr (CONCAT) {
    const float* ap2 = A2 + (long)asrc * lda2 + khalf;
#pragma unroll
    for (int k0 = 64; k0 < K; k0 += 4) {
      v2f a = *(const v2f*)(ap2 + (k0 - 64));
      v2f b0, b1;
      b0.x = wp0[k0 * LDW];     b0.y = wp0[k0 * LDW + LDW];
      b1.x = wp1[k0 * LDW];     b1.y = wp1[k0 * LDW + LDW];
      acc0 = __builtin_amdgcn_wmma_f32_16x16x4_f32(false, a, false, b0,
                                                   (short)0, acc0, false, false);
      acc1 = __builtin_amdgcn_wmma_f32_16x16x4_f32(false, a, false, b1,
                                                   (short)0, acc1, false, false);
    }
  }

  // epilogue: D rows: lanes0-15 -> M=v, lanes16-31 -> M=v+8
  int rbase = rowTile * 16 + (lane >> 4) * 8;
  float bv0 = bias[col0];
  float bv1 = bias[c1];
  bool ok1 = (NOUT % 32 == 0) ? true : (col1 < NOUT);  // folds for 64/128
#pragma unroll
  for (int v = 0; v < 8; ++v) {
    int r = rbase + v;
    float x0 = acc0[v] + bv0;
    float x1 = acc1[v] + bv1;
    if constexpr (ACT == 1) {
      x0 = fmaxf(x0, 0.f);
      x1 = fmaxf(x1, 0.f);
    } else if constexpr (ACT == 2) {
      float g0 = 1.f / (1.f + __expf(-x0));
      float g1 = 1.f / (1.f + __expf(-x1));
      x0 = fmaxf(gate_src[(long)r * ldg + col0] * g0, 0.f);
      if (ok1) x1 = fmaxf(gate_src[(long)r * ldg + col1] * g1, 0.f);
    }
    float* orow = Out + (long)r * ldo + ocol;
    orow[col0] = x0;
    if (ok1) orow[col1] = x1;
  }
}

template<int K, int LDW, int NOUT, int ACT, bool CONCAT, bool GATHER>
static void launch_gemm(hipStream_t s,
    const float* A, int lda, const float* A2, int lda2, const int* gidx,
    const float* W, const float* bias,
    float* Out, int ldo, int ocol, const float* gate, int ldg, int Nrows)
{
  constexpr int COLPAIRS = ((NOUT + 15) / 16) / 2;
  long waves = (long)(Nrows / 16) * COLPAIRS;
  int blocks = (int)((waves + 7) / 8);   // 8 waves per 256-thread block
  gemm16_wmma<K, LDW, NOUT, ACT, CONCAT, GATHER><<<blocks, 256, 0, s>>>(
      A, lda, A2, lda2, gidx, W, bias, Out, ldo, ocol, gate, ldg, Nrows);
}

// -------------------- voxel majority vote --------------------
__global__ __launch_bounds__(256) void scatter_counts(
    const int* __restrict__ labels, const int* __restrict__ inv,
    int n, unsigned* __restrict__ counts)
{
  int i = blockIdx.x * 256 + threadIdx.x;
  if (i >= n) return;
  atomicAdd(&counts[(long)inv[i] * NCLS + labels[i]], 1u);
}

__global__ __launch_bounds__(256) void argmax_vox(
    const unsigned* __restrict__ counts, int n, int* __restrict__ voxlab)
{
  int i = blockIdx.x * 256 + threadIdx.x;
  if (i >= n) return;
  const unsigned* c = counts + (long)i * NCLS;
  unsigned bv = c[0]; int bi = 0;
  for (int k = 1; k < NCLS; ++k) if (c[k] > bv) { bv = c[k]; bi = k; }
  voxlab[i] = bi;
}

__global__ __launch_bounds__(256) void gather_idx(
    const int* __restrict__ inv, const int* __restrict__ p2img,
    int n, int* __restrict__ idx)
{
  int i = blockIdx.x * 256 + threadIdx.x;
  if (i >= n) return;
  idx[i] = inv[p2img[i]];
}

// -------------------- cross-entropy (ignore_index=0) --------------------
__global__ __launch_bounds__(256) void ce_kernel(
    const float* __restrict__ logits, const int* __restrict__ labels,
    int n, float* __restrict__ partials)
{
  int i = blockIdx.x * 256 + threadIdx.x;
  float nll = 0.f, valid = 0.f;
  if (i < n) {
    int lab = labels[i];
    if (lab != 0) {
      const float* x = logits + (long)i * NCLS;
      float mx = x[0];
      for (int c = 1; c < NCLS; ++c) mx = fmaxf(mx, x[c]);
      float s = 0.f;
      for (int c = 0; c < NCLS; ++c) s += __expf(x[c] - mx);
      nll = (mx + __logf(s)) - x[lab];
      valid = 1.f;
    }
  }
  __shared__ float sa[256], sb[256];
  int t = threadIdx.x;
  sa[t] = nll; sb[t] = valid; __syncthreads();
  for (int off = 128; off > 0; off >>= 1) {
    if (t < off) { sa[t] += sa[t + off]; sb[t] += sb[t + off]; }
    __syncthreads();
  }
  if (t == 0) { partials[2 * blockIdx.x] = sa[0]; partials[2 * blockIdx.x + 1] = sb[0]; }
}

// -------------------- KL(q || p) term --------------------
__global__ __launch_bounds__(256) void kl_kernel(
    const float* __restrict__ pts_pred, const float* __restrict__ fuse_pred,
    const int* __restrict__ idx, int n, float* __restrict__ partials)
{
  int i = blockIdx.x * 256 + threadIdx.x;
  float s = 0.f, cnt = 0.f;
  if (i < n) {
    const float* fp = fuse_pred + (long)i * NCLS;
    const float* pp = pts_pred + (long)idx[i] * NCLS;
    float mf = fp[0], mp = pp[0];
    for (int c = 1; c < NCLS; ++c) { mf = fmaxf(mf, fp[c]); mp = fmaxf(mp, pp[c]); }
    float sf = 0.f, sp = 0.f;
    for (int c = 0; c < NCLS; ++c) { sf += __expf(fp[c] - mf); sp += __expf(pp[c] - mp); }
    float lf = mf + __logf(sf), lp = mp + __logf(sp);
    for (int c = 0; c < NCLS; ++c) {
      float lq = fp[c] - lf;
      s += __expf(lq) * (lq - (pp[c] - lp));
    }
    cnt = (float)NCLS;
  }
  __shared__ float sa[256], sb[256];
  int t = threadIdx.x;
  sa[t] = s; sb[t] = cnt; __syncthreads();
  for (int off = 128; off > 0; off >>= 1) {
    if (t < off) { sa[t] += sa[t + off]; sb[t] += sb[t + off]; }
    __syncthreads();
  }
  if (t == 0) { partials[2 * blockIdx.x] = sa[0]; partials[2 * blockIdx.x + 1] = sb[0]; }
}

__global__ __launch_bounds__(256) void finalize_ratio(
    const float* __restrict__ partials, int nblocks, float scale,
    float* __restrict__ slot)
{
  __shared__ float sa[256], sb[256];
  int t = threadIdx.x;
  float a = 0.f, b = 0.f;
  for (int j = t; j < nblocks; j += 256) { a += partials[2 * j]; b += partials[2 * j + 1]; }
  sa[t] = a; sb[t] = b; __syncthreads();
  for (int off = 128; off > 0; off >>= 1) {
    if (t < off) { sa[t] += sa[t + off]; sb[t] += sb[t + off]; }
    __syncthreads();
  }
  if (t == 0) *slot = scale * sa[0] / fmaxf(sb[0], 1.f);
}

// -------------------- Lovasz-softmax via binned telescoping --------------------
// Per valid point, per class: err=|fg-p_c| -> bin; integer atomics keep it
// deterministic. Error sums kept in Q32.32 fixed point (u64 atomic).
__global__ __launch_bounds__(256) void lovasz_hist(
    const float* __restrict__ logits, const int* __restrict__ labels, int n,
    unsigned* __restrict__ hn, unsigned* __restrict__ hf,
    unsigned long long* __restrict__ he)
{
  int i = blockIdx.x * 256 + threadIdx.x;
  if (i >= n) return;
  int lab = labels[i];
  if (lab == 0) return;  // err==0 tail: contributes nothing to dot(e_s, grad)
  const float* x = logits + (long)i * NCLS;
  float mx = x[0];
  for (int c = 1; c < NCLS; ++c) mx = fmaxf(mx, x[c]);
  float s = 0.f;
  for (int c = 0; c < NCLS; ++c) s += __expf(x[c] - mx);
  float invs = 1.f / s;
  for (int c = 0; c < NCLS; ++c) {
    float p = __expf(x[c] - mx) * invs;
    float fg = (c == lab) ? 1.f : 0.f;
    float err = fabsf(fg - p);
    int b = (int)(err * (float)NBINS);
    if (b >= NBINS) b = NBINS - 1;
    atomicAdd(&hn[c * NBINS + b], 1u);
    if (fg > 0.f) atomicAdd(&hf[c * NBINS + b], 1u);
    atomicAdd(&he[c * NBINS + b], (unsigned long long)((double)err * 4294967296.0));
  }
}

__device__ float blk_scan_incl(float v, float* sh)
{
  int t = threadIdx.x;
  __syncthreads();
  int pin = 0;
  sh[t] = v; __syncthreads();
  for (int off = 1; off < 256; off <<= 1) {
    int pout = pin ^ 1;
    float nv = sh[pin * 256 + t];
    if (t >= off) nv += sh[pin * 256 + t - off];
    sh[pout * 256 + t] = nv;
    __syncthreads();
    pin = pout;
  }
  return sh[pin * 256 + t];
}

__global__ __launch_bounds__(256) void lovasz_scan(
    const unsigned* __restrict__ hn, const unsigned* __restrict__ hf,
    const unsigned long long* __restrict__ he, float* __restrict__ cls_res)
{
  int c = blockIdx.x;
  const unsigned* n = hn + c * NBINS;
  const unsigned* f = hf + c * NBINS;
  const unsigned long long* e = he + c * NBINS;
  int t = threadIdx.x;
  __shared__ float sh[512];
  __shared__ float s_gts, s_cn, s_cf;

  float fs = 0.f;
  for (int j = t; j < NBINS; j += 256) fs += (float)f[j];
  sh[t] = fs; __syncthreads();
  for (int off = 128; off > 0; off >>= 1) { if (t < off) sh[t] += sh[t + off]; __syncthreads(); }
  if (t == 0) { s_gts = sh[0]; s_cn = 0.f; s_cf = 0.f; }
  __syncthreads();
  float gts = s_gts;

  float total = 0.f;
  for (int base = 0; base < NBINS; base += 256) {
    int bin = NBINS - 1 - (base + t);   // descending error order
    float nj = (float)n[bin];
    float fj = (float)f[bin];
    float sn = blk_scan_incl(nj, sh);
    float sf = blk_scan_incl(fj, sh);
    float cn = s_cn + sn, cf = s_cf + sf;
    float cnp = cn - nj, cfp = cf - fj;
    if (nj > 0.f) {
      float jac  = 1.f - (gts - cf ) / fmaxf(gts + cn  - cf , 1.f);
      float jacp = 1.f - (gts - cfp) / fmaxf(gts + cnp - cfp, 1.f);
      float em = (float)((double)e[bin] * (1.0 / 4294967296.0) / (double)nj);
      total += em * (jac - jacp);     // telescoped within equal-err bin
    }
    __syncthreads();
    if (t == 255) { s_cn += sn; s_cf += sf; }
    __syncthreads();
  }
  sh[t] = total; __syncthreads();
  for (int off = 128; off > 0; off >>= 1) { if (t < off) sh[t] += sh[t + off]; __syncthreads(); }
  if (t == 0) {
    float present = (gts > 0.f) ? 1.f : 0.f;
    cls_res[2 * c]     = sh[0] * present;
    cls_res[2 * c + 1] = present;
  }
}

__global__ void lovasz_combine(const float* __restrict__ cls_res, float scale,
                               float* __restrict__ slot)
{
  if (threadIdx.x == 0 && blockIdx.x == 0) {
    float ls = 0.f, ps = 0.f;
    for (int c = 0; c < NCLS; ++c) { ls += cls_res[2 * c]; ps += cls_res[2 * c + 1]; }
    *slot = scale * ls / fmaxf(ps, 1.f);
  }
}

__global__ void sum_slots(const float* __restrict__ slots, int n, float* __restrict__ out)
{
  if (threadIdx.x == 0 && blockIdx.x == 0) {
    float s = 0.f;
    for (int i = 0; i < n; ++i) s += slots[i];
    out[0] = s;
  }
}

// ---------------------------------------------------------------------------
extern "C" void kernel_launch(void* const* d_in, const int* in_sizes, int n_in,
                              void* d_out, int out_size, void* d_ws, size_t ws_size,
                              hipStream_t stream)
{
  const float* pts_feats = (const float*)d_in[0];   // [NSC, NVOX, 64]
  const float* img_feats = (const float*)d_in[1];   // [NSC, MPTS, 64]
  const float* P         = (const float*)d_in[2];   // flat params (tree order)
  const int*   coors_inv = (const int*)d_in[3];     // [NSC, NPTS]
  const int*   p2img     = (const int*)d_in[4];     // [MPTS]
  const int*   labels    = (const int*)d_in[5];     // [NPTS]
  const int*   img_label = (const int*)d_in[6];     // [MPTS]
  float* out = (float*)d_out;

  // params flat layout (jax pytree: dict keys sorted alphabetically)
  const int CLS3D = 0;         // per scale 10900: W1(64x128), b1, W2(128x20), b2
  const int CLSFUSE = 43600;
  const int FC1 = 87200;       // per scale 8256: W(128x64), b
  const int FC2 = 120224;      // per scale 4160: W(64x64), b
  const int FINAL_ = 136864;   // W1(256x128) b1 W2(128x20) b2
  const int LEANER = 172340;   // per scale 4160

  size_t off = 0;
  auto alloc = [&](size_t bytes) -> void* {
    void* p = (char*)d_ws + off;
    off += (bytes + 255) & ~(size_t)255;
    return p;
  };
  float* fuse_cat  = (float*)alloc((size_t)MPTS * 256 * 4); // persists all scales
  float* big0      = (float*)alloc((size_t)MPTS * 128 * 4); // hidden scratch
  float* pts_pred  = (float*)alloc((size_t)NVOX * NCLS * 4);
  float* fuse_pred = (float*)alloc((size_t)MPTS * NCLS * 4);
  float* feat_lrn  = (float*)alloc((size_t)MPTS * 64 * 4);
  float* feat_cat  = (float*)alloc((size_t)MPTS * 64 * 4);
  float* partials  = (float*)alloc((size_t)2 * 4096 * 4);
  float* cls_res   = (float*)alloc(64 * 4);
  float* slots     = (float*)alloc(64 * 4);
  unsigned* counts = (unsigned*)alloc((size_t)NVOX * NCLS * 4);
  int* voxlab      = (int*)alloc((size_t)NVOX * 4);
  int* idxbuf      = (int*)alloc((size_t)MPTS * 4);
  unsigned* hn     = (unsigned*)alloc((size_t)NCLS * NBINS * 4);
  unsigned* hf     = (unsigned*)alloc((size_t)NCLS * NBINS * 4);
  unsigned long long* he = (unsigned long long*)alloc((size_t)NCLS * NBINS * 8);

  const size_t HIST4 = (size_t)NCLS * NBINS * 4;
  const int nbV = (NVOX + 255) / 256;
  const int nbM = (MPTS + 255) / 256;
  const int nbP = (NPTS + 255) / 256;

  for (int i = 0; i < NSC; ++i) {
    const float* pf  = pts_feats + (size_t)i * NVOX * HDIM;
    const float* imf = img_feats + (size_t)i * MPTS * HDIM;
    const int*   inv = coors_inv + (size_t)i * NPTS;

    const float* c3W1 = P + CLS3D + i * 10900;
    const float* c3b1 = c3W1 + 8192;
    const float* c3W2 = c3W1 + 8320;
    const float* c3b2 = c3W1 + 10880;
    const float* cfW1 = P + CLSFUSE + i * 10900;
    const float* cfb1 = cfW1 + 8192;
    const float* cfW2 = cfW1 + 8320;
    const float* cfb2 = cfW1 + 10880;
    const float* f1W = P + FC1 + i * 8256;  const float* f1b = f1W + 8192;
    const float* f2W = P + FC2 + i * 4160;  const float* f2b = f2W + 4096;
    const float* lnW = P + LEANER + i * 4160; const float* lnb = lnW + 4096;

    // 3D branch: pts_pred_full = relu(pf@W1+b1)@W2+b2
    launch_gemm<64, 128, 128, 1, false, false>(stream, pf, 64, 0, 0, 0,
        c3W1, c3b1, big0, 128, 0, 0, 0, NVOX);
    launch_gemm<128, 20, 20, 0, false, false>(stream, big0, 128, 0, 0, 0,
        c3W2, c3b2, pts_pred, 20, 0, 0, 0, NVOX);

    // voxel majority-vote labels
    hipMemsetAsync(counts, 0, (size_t)NVOX * NCLS * 4, stream);
    scatter_counts<<<nbP, 256, 0, stream>>>(labels, inv, NPTS, counts);
    argmax_vox<<<nbV, 256, 0, stream>>>(counts, NVOX, voxlab);

    // composed gather index
    gather_idx<<<nbM, 256, 0, stream>>>(inv, p2img, MPTS, idxbuf);

    // fusion path
    launch_gemm<64, 64, 64, 1, false, true>(stream, pf, 64, 0, 0, idxbuf,
        lnW, lnb, feat_lrn, 64, 0, 0, 0, MPTS);
    launch_gemm<128, 64, 64, 0, true, false>(stream, imf, 64, feat_lrn, 64, 0,
        f1W, f1b, feat_cat, 64, 0, 0, 0, MPTS);
    launch_gemm<64, 64, 64, 2, false, false>(stream, feat_cat, 64, 0, 0, 0,
        f2W, f2b, fuse_cat, 256, 64 * i, feat_cat, 64, MPTS); // fused gate
    launch_gemm<64, 128, 128, 1, false, false>(stream, fuse_cat + 64 * i, 256, 0, 0, 0,
        cfW1, cfb1, big0, 128, 0, 0, 0, MPTS);
    launch_gemm<128, 20, 20, 0, false, false>(stream, big0, 128, 0, 0, 0,
        cfW2, cfb2, fuse_pred, 20, 0, 0, 0, MPTS);

    // seg loss (3D): CE + lovasz
    ce_kernel<<<nbV, 256, 0, stream>>>(pts_pred, voxlab, NVOX, partials);
    finalize_ratio<<<1, 256, 0, stream>>>(partials, nbV, 1.f, slots + 5 * i + 0);
    hipMemsetAsync(hn, 0, HIST4, stream);
    hipMemsetAsync(hf, 0, HIST4, stream);
    hipMemsetAsync(he, 0, HIST4 * 2, stream);
    lovasz_hist<<<nbV, 256, 0, stream>>>(pts_pred, voxlab, NVOX, hn, hf, he);
    lovasz_scan<<<NCLS, 256, 0, stream>>>(hn, hf, he, cls_res);
    lovasz_combine<<<1, 1, 0, stream>>>(cls_res, 1.f, slots + 5 * i + 1);

    // seg loss (2D fuse) * (LAMBDA_SEG2D/NS)
    ce_kernel<<<nbM, 256, 0, stream>>>(fuse_pred, img_label, MPTS, partials);
    finalize_ratio<<<1, 256, 0, stream>>>(partials, nbM, 0.25f, slots + 5 * i + 2);
    hipMemsetAsync(hn, 0, HIST4, stream);
    hipMemsetAsync(hf, 0, HIST4, stream);
    hipMemsetAsync(he, 0, HIST4 * 2, stream);
    lovasz_hist<<<nbM, 256, 0, stream>>>(fuse_pred, img_label, MPTS, hn, hf, he);
    lovasz_scan<<<NCLS, 256, 0, stream>>>(hn, hf, he, cls_res);
    lovasz_combine<<<1, 1, 0, stream>>>(cls_res, 0.25f, slots + 5 * i + 3);

    // KL * (LAMBDA_XM/NS)
    kl_kernel<<<nbM, 256, 0, stream>>>(pts_pred, fuse_pred, idxbuf, MPTS, partials);
    finalize_ratio<<<1, 256, 0, stream>>>(partials, nbM, 0.0125f, slots + 5 * i + 4);
  }

  // final classifier on concatenated fuse feats
  launch_gemm<256, 128, 128, 1, false, false>(stream, fuse_cat, 256, 0, 0, 0,
      P + FINAL_, P + FINAL_ + 32768, big0, 128, 0, 0, 0, MPTS);
  launch_gemm<128, 20, 20, 0, false, false>(stream, big0, 128, 0, 0, 0,
      P + FINAL_ + 32896, P + FINAL_ + 35456, fuse_pred, 20, 0, 0, 0, MPTS);
  ce_kernel<<<nbM, 256, 0, stream>>>(fuse_pred, img_label, MPTS, partials);
  finalize_ratio<<<1, 256, 0, stream>>>(partials, nbM, 1.f, slots + 20);
  hipMemsetAsync(hn, 0, HIST4, stream);
  hipMemsetAsync(hf, 0, HIST4, stream);
  hipMemsetAsync(he, 0, HIST4 * 2, stream);
  lovasz_hist<<<nbM, 256, 0, stream>>>(fuse_pred, img_label, MPTS, hn, hf, he);
  lovasz_scan<<<NCLS, 256, 0, stream>>>(hn, hf, he, cls_res);
  lovasz_combine<<<1, 1, 0, stream>>>(cls_res, 1.f, slots + 21);

  sum_slots<<<1, 1, 0, stream>>>(slots, 22, out);
}